// ROSA_1bit_LAYER_47665547051800
// MI455X (gfx1250) — compile-verified
//
#include <hip/hip_runtime.h>

#define Bn 4
#define Tn 512
#define Cn 32

typedef float v2f __attribute__((ext_vector_type(2)));
typedef float v8f __attribute__((ext_vector_type(8)));

// One block per batch b: 32 waves (wave32), wave y owns column c = y.
// Phase 1: coalesced sign load -> LDS bytes.
// Phase 2: per-column ballot-pack into 16 x u32 bit-words (LDS).
// Phase 3: per-wave diagonal run-length scan over t (16 lags/lane),
//          wave32 shfl_xor max-reduce with tie-break toward smallest k.
// Phase 4: WMMA epilogue: Out = S0*diag(e0) + S1*diag(e1) via
//          v_wmma_f32_16x16x4_f32 (exact fp32).
__global__ __launch_bounds__(1024) void rosa_kernel(const float* __restrict__ x,
                                                    const float* __restrict__ emb0,
                                                    const float* __restrict__ emb1,
                                                    float* __restrict__ out)
{
    __shared__ signed char lds_sc[Tn][Cn];     // phase 1: bits, phase 3: idx
    __shared__ unsigned    lds_words[Cn][16];  // packed bit-words per column

    const int lane = threadIdx.x;   // 0..31
    const int c    = threadIdx.y;   // 0..31 == wave id
    const int b    = blockIdx.x;
    const int tid  = c * 32 + lane;

    const float* __restrict__ xb = x + b * Tn * Cn;
    __builtin_prefetch(xb + tid, 0, 1);   // global_prefetch_b8

    // ---- Phase 1: coalesced sign loads into LDS bytes ----
#pragma unroll
    for (int e = tid; e < Tn * Cn; e += 1024) {
        lds_sc[e >> 5][e & 31] = (xb[e] > 0.0f) ? (signed char)1 : (signed char)0;
    }
    __syncthreads();

    // ---- Phase 2: wave32 ballot pack: column c -> 16 u32 words ----
    unsigned myword = 0;
#pragma unroll
    for (int j = 0; j < 16; ++j) {
        unsigned w = (unsigned)__ballot(lds_sc[j * 32 + lane][c] != 0);
        if (lane == j) myword = w;
    }
    if (lane < 16) lds_words[c][lane] = myword;
    __syncthreads();   // words ready; lds_sc reusable for idx results

    // ---- Phase 3: diagonal run scan; lane handles k = 1+lane+32j ----
    int run[16];
#pragma unroll
    for (int j = 0; j < 16; ++j) run[j] = 0;

    for (int w = 0; w < 16; ++w) {
        const unsigned bw = lds_words[c][w];
        for (int i = 0; i < 32; ++i) {
            const int t  = (w << 5) + i;
            const int bt = (int)((bw >> i) & 1u);
            int best = 0;
#pragma unroll
            for (int j = 0; j < 16; ++j) {
                const int k = 1 + lane + (j << 5);
                const int s = t - k;
                int m = 0;
                if (s >= 0) {
                    const unsigned sw = lds_words[c][s >> 5];
                    m = ((int)((sw >> (s & 31)) & 1u) == bt);
                }
                run[j] = m ? (run[j] + 1) : 0;
                // key packs (runlen, 512-k): max => longest run, tie => smallest k
                const int key = (run[j] << 10) | (512 - k);
                best = (key > best) ? key : best;
            }
            // wave32 max reduction
#pragma unroll
            for (int off = 16; off > 0; off >>= 1) {
                const int o = __shfl_xor(best, off, 32);
                best = (o > best) ? o : best;
            }
            int idxv = -1;
            const int L = best >> 10;
            if (L > 0) {
                const int kstar = 512 - (best & 1023);
                const int nxt   = t - kstar + 1;          // in [1, t]
                idxv = (int)((lds_words[c][nxt >> 5] >> (nxt & 31)) & 1u);
            }
            if (lane == 0) lds_sc[t][c] = (signed char)idxv;
        }
    }
    __syncthreads();

    // ---- Phase 4: WMMA epilogue. Out_tile = S0*diag(e0) + S1*diag(e1). ----
    // 64 tiles (32 t-tiles x 2 c-tiles) over 32 waves -> 2 tiles/wave.
    const int mr   = lane & 15;   // A row M / B col N / D col N
    const int half = lane >> 4;   // K half selector / D row offset
#pragma unroll
    for (int it = 0; it < 2; ++it) {
        const int tile = c * 2 + it;
        const int t0   = (tile >> 1) << 4;
        const int c0   = (tile & 1) << 4;
        const float e0v = emb0[c0 + mr];
        const float e1v = emb1[c0 + mr];
        v8f acc = {};
#pragma unroll
        for (int kk = 0; kk < 16; kk += 4) {
            const int ka = kk + 2 * half;                 // global K for v=0
            const signed char i0 = lds_sc[t0 + mr][c0 + ka];
            const signed char i1 = lds_sc[t0 + mr][c0 + ka + 1];
            v2f a0 = { (i0 == 0) ? 1.0f : 0.0f, (i1 == 0) ? 1.0f : 0.0f };
            v2f a1 = { (i0 == 1) ? 1.0f : 0.0f, (i1 == 1) ? 1.0f : 0.0f };
            v2f b0 = { (mr == ka) ? e0v : 0.0f, (mr == ka + 1) ? e0v : 0.0f };
            v2f b1 = { (mr == ka) ? e1v : 0.0f, (mr == ka + 1) ? e1v : 0.0f };
            acc = __builtin_amdgcn_wmma_f32_16x16x4_f32(
                false, a0, false, b0, (short)0, acc, false, false);
            acc = __builtin_amdgcn_wmma_f32_16x16x4_f32(
                false, a1, false, b1, (short)0, acc, false, false);
        }
        float* __restrict__ outb = out + b * Tn * Cn;
#pragma unroll
        for (int r = 0; r < 8; ++r) {
            outb[(t0 + r + 8 * half) * Cn + (c0 + mr)] = acc[r];
        }
    }
}

extern "C" void kernel_launch(void* const* d_in, const int* in_sizes, int n_in,
                              void* d_out, int out_size, void* d_ws, size_t ws_size,
                              hipStream_t stream) {
    const float* x  = (const float*)d_in[0];
    const float* e0 = (const float*)d_in[1];
    const float* e1 = (const float*)d_in[2];
    float* out = (float*)d_out;
    (void)in_sizes; (void)n_in; (void)out_size; (void)d_ws; (void)ws_size;
    dim3 grid(Bn);
    dim3 block(32, 32);
    rosa_kernel<<<grid, block, 0, stream>>>(x, e0, e1, out);
}